// BerTII_50251117363580
// MI455X (gfx1250) — compile-verified
//
#include <hip/hip_runtime.h>
#include <math.h>

#define VOCAB 200000
#define PDIM  1000
#define BATCH 64
#define SEQL  2048

typedef float v2f __attribute__((ext_vector_type(2)));
typedef float v4f __attribute__((ext_vector_type(4)));
typedef float v8f __attribute__((ext_vector_type(8)));

// ---------------------------------------------------------------------------
// Kernel 1:  s[v] = dot(E[v, :], W)   for v in [0, VOCAB)
//
// One wave32 handles a 16-row tile of E via V_WMMA_F32_16X16X4_F32:
//   A (16x4 f32, documented layout): lane l%16 = row m, lane half selects
//     K={0,1}/{2,3} -> each lane streams a float2 of its row (NT-hinted).
//   B (4x16 f32): W chunk broadcast to ALL 16 columns -> D[m][n] is the same
//     partial dot product for every n, so correctness does not depend on the
//     (undocumented) B-column / D-column lane mapping, only on the K map,
//     which mirrors the documented A layout.
//   D (16x16 f32, documented layout): lane 0 holds rows 0..7 (VGPRs 0..7),
//     lane 16 holds rows 8..15.
// W (4 KB) is cached in LDS so the VMEM pipe carries only the 800 MB E
// stream; B operands come from the independent DS pipe.
// 200000 rows / 16 = 12500 tiles = 3125 blocks x 4 waves.
// ---------------------------------------------------------------------------
__global__ __launch_bounds__(128)
void emb_gemv_wmma(const float* __restrict__ E, const float* __restrict__ W,
                   float* __restrict__ s_out) {
  __shared__ float lds_w[PDIM];
  for (int i = threadIdx.x; i < PDIM; i += 128) lds_w[i] = W[i];
  __syncthreads();

  const int lane  = threadIdx.x & 31;
  const int wave  = threadIdx.x >> 5;
  const int tile  = blockIdx.x * 4 + wave;
  const int m     = lane & 15;   // row within tile
  const int khalf = lane >> 4;   // 0 -> K={0,1}, 1 -> K={2,3}

  const float* rowp = E + (size_t)(tile * 16 + m) * PDIM + 2 * khalf;
  const float* wp   = &lds_w[2 * khalf];

  v8f acc0 = {};
  v8f acc1 = {};

  for (int kb = 0; kb < PDIM; kb += 8) {
    // 800 MB single-use stream: non-temporal so it doesn't wash s[] out of L2.
    v2f a0 = __builtin_nontemporal_load((const v2f*)(rowp + kb));
    v2f a1 = __builtin_nontemporal_load((const v2f*)(rowp + kb + 4));
    // W chunk from LDS (broadcast read, conflict-free), replicated across all
    // B columns.
    v2f b0 = *(const v2f*)(wp + kb);
    v2f b1 = *(const v2f*)(wp + kb + 4);
    // 8 args: (neg_a, A, neg_b, B, c_mod, C, reuse_a, reuse_b)
    acc0 = __builtin_amdgcn_wmma_f32_16x16x4_f32(false, a0, false, b0,
                                                 (short)0, acc0, false, false);
    acc1 = __builtin_amdgcn_wmma_f32_16x16x4_f32(false, a1, false, b1,
                                                 (short)0, acc1, false, false);
  }

  float acc[8];
#pragma unroll
  for (int i = 0; i < 8; ++i) acc[i] = acc0[i] + acc1[i];

  if (m == 0) {  // lane 0: rows 0..7, lane 16: rows 8..15 (D layout, any column)
    float* outp = s_out + tile * 16 + 8 * khalf;
    v4f lo = {acc[0], acc[1], acc[2], acc[3]};
    v4f hi = {acc[4], acc[5], acc[6], acc[7]};
    *(v4f*)(outp)     = lo;
    *(v4f*)(outp + 4) = hi;
  }
}

// ---------------------------------------------------------------------------
// Kernel 2: logits[b] = sigmoid( (1/L) * sum_{l<len_b} s[tok[b,l]] + bias )
// One block per batch row; s[] (800 KB) is L2-resident after kernel 1.
// ---------------------------------------------------------------------------
__global__ __launch_bounds__(256)
void pool_head(const int* __restrict__ tokens, const int* __restrict__ lengths,
               const float* __restrict__ s, const float* __restrict__ bias_ptr,
               float* __restrict__ out) {
  __shared__ float red[256];
  const int bi  = blockIdx.x;
  const int t   = threadIdx.x;
  const int len = lengths[bi];
  const int* tok = tokens + bi * SEQL;

  float sum = 0.0f;
  for (int l = t; l < len; l += 256) sum += s[tok[l]];

  red[t] = sum;
  __syncthreads();
#pragma unroll
  for (int off = 128; off > 0; off >>= 1) {
    if (t < off) red[t] += red[t + off];
    __syncthreads();
  }

  if (t == 0) {
    float x = red[0] * (1.0f / (float)SEQL) + bias_ptr[0];
    out[bi] = 1.0f / (1.0f + expf(-x));
  }
}

extern "C" void kernel_launch(void* const* d_in, const int* in_sizes, int n_in,
                              void* d_out, int out_size, void* d_ws, size_t ws_size,
                              hipStream_t stream) {
  (void)in_sizes; (void)n_in; (void)out_size; (void)ws_size;

  const int*   tokens  = (const int*)d_in[0];    // [B, L]
  const int*   lengths = (const int*)d_in[1];    // [B]
  const float* E       = (const float*)d_in[2];  // [VOCAB, P]
  const float* W       = (const float*)d_in[3];  // [1, P]
  const float* b       = (const float*)d_in[4];  // [1]
  float* out = (float*)d_out;                    // [B]
  float* s   = (float*)d_ws;                     // VOCAB floats = 800 KB scratch

  emb_gemv_wmma<<<VOCAB / 16 / 4, 128, 0, stream>>>(E, W, s);
  pool_head<<<BATCH, 256, 0, stream>>>(tokens, lengths, s, b, out);
}